// LocalGrouper_9758165697099
// MI455X (gfx1250) — compile-verified
//
#include <hip/hip_runtime.h>

#define NB   16      // batches
#define NPT  4096    // points
#define NC   128     // channels
#define NS   1024    // fps samples
#define NK   24      // neighbors
#define EPSV 1e-5f

typedef __attribute__((ext_vector_type(2))) float v2f;
typedef __attribute__((ext_vector_type(8))) float v8f;

// ---------------- Kernel 1: farthest point sampling, one block per batch ----
// 256 threads, 16 points/thread in registers. Per step: shuffle-based wave
// argmax (no barriers) + single block barrier over 8 wave leaders
// (double-buffered LDS removes the WAR barrier).
__global__ __launch_bounds__(256) void fps_kernel(const float* __restrict__ xyz,
                                                  int* __restrict__ fps_idx) {
    const int b    = blockIdx.x;
    const int tid  = threadIdx.x;
    const int lane = tid & 31;
    const int wave = tid >> 5;
    __shared__ float sval[2][8];
    __shared__ int   sidx[2][8];
    float px[16], py[16], pz[16], md[16];
#pragma unroll
    for (int j = 0; j < 16; ++j) {
        int n = tid + j * 256;
        const float* q = xyz + ((size_t)b * NPT + n) * 3;
        px[j] = q[0]; py[j] = q[1]; pz[j] = q[2];
        md[j] = 3.4e38f;
    }
    int last = 0;
    for (int s = 0; s < NS; ++s) {
        if (tid == 0) fps_idx[b * NS + s] = last;
        const float* q = xyz + ((size_t)b * NPT + last) * 3;
        float lx = q[0], ly = q[1], lz = q[2];
        float bv = -1.f; int bi = 0x7FFFFFFF;
#pragma unroll
        for (int j = 0; j < 16; ++j) {
            float dx = px[j] - lx, dy = py[j] - ly, dz = pz[j] - lz;
            float d = dx * dx + dy * dy + dz * dz;
            md[j] = fminf(md[j], d);
            int n = tid + j * 256;
            if (md[j] > bv || (md[j] == bv && n < bi)) { bv = md[j]; bi = n; }
        }
        // intra-wave argmax (tie -> smaller index), wave32 shuffles
#pragma unroll
        for (int off = 16; off > 0; off >>= 1) {
            float ov = __shfl_xor(bv, off, 32);
            int   oi = __shfl_xor(bi, off, 32);
            if (ov > bv || (ov == bv && oi < bi)) { bv = ov; bi = oi; }
        }
        int par = s & 1;
        if (lane == 0) { sval[par][wave] = bv; sidx[par][wave] = bi; }
        __syncthreads();
        float mbv = sval[par][0]; int mbi = sidx[par][0];
#pragma unroll
        for (int w = 1; w < 8; ++w) {
            float ov = sval[par][w]; int oi = sidx[par][w];
            if (ov > mbv || (ov == mbv && oi < mbi)) { mbv = ov; mbi = oi; }
        }
        last = mbi;
    }
}

// ---------------- Kernel 2: gather new_xyz into output head ----------------
__global__ void gather_xyz_kernel(const float* __restrict__ xyz,
                                  const int* __restrict__ fps_idx,
                                  float* __restrict__ out) {
    int i = blockIdx.x * blockDim.x + threadIdx.x;
    if (i >= NB * NS) return;
    int b = i / NS;
    int idx = fps_idx[i];
    const float* q = xyz + ((size_t)b * NPT + idx) * 3;
    out[i * 3 + 0] = q[0];
    out[i * 3 + 1] = q[1];
    out[i * 3 + 2] = q[2];
}

// --------- Kernel 3: WMMA distance panel (16 x 4096 in LDS) + top-K --------
// grid = NB * (NS/16) blocks, 256 threads (8 waves), 256 KB dynamic LDS.
__global__ __launch_bounds__(256) void knn_kernel(const float* __restrict__ xyz,
                                                  const int* __restrict__ fps_idx,
                                                  int* __restrict__ knn_idx) {
    extern __shared__ float dist[];            // 16 rows x 4096 cols of f32
    const int b     = blockIdx.x >> 6;         // 64 tiles of 16 centroids per batch
    const int stile = blockIdx.x & 63;
    const int tid   = threadIdx.x;
    const int wave  = tid >> 5;
    const int lane  = tid & 31;
    const int m     = lane & 15;
    const int half  = lane >> 4;               // A/B lane halves hold K={0,1} / {2,3}

    // A matrix (16x4): [-2x, -2y, -2z, ||s||^2] per centroid row
    int sIdx = fps_idx[b * NS + stile * 16 + m];
    const float* cq = xyz + ((size_t)b * NPT + sIdx) * 3;
    float cx = cq[0], cy = cq[1], cz = cq[2];
    float s2 = cx * cx + cy * cy + cz * cz;
    v2f A;
    A[0] = half ? (-2.f * cz) : (-2.f * cx);
    A[1] = half ? s2          : (-2.f * cy);

    // Each wave owns column tiles t = wave, wave+8, ... (32 tiles of 16 cols)
    for (int t = wave; t < NPT / 16; t += 8) {
        int n = t * 16 + m;
        const float* pq = xyz + ((size_t)b * NPT + n) * 3;
        float pxx = pq[0], pyy = pq[1], pzz = pq[2];
        float n2 = pxx * pxx + pyy * pyy + pzz * pzz;
        v2f Bm;                               // B matrix (4x16): [x; y; z; 1]
        Bm[0] = half ? pzz  : pxx;
        Bm[1] = half ? 1.0f : pyy;
        v8f Cm;                               // C = ||n||^2 broadcast down the column
#pragma unroll
        for (int r = 0; r < 8; ++r) Cm[r] = n2;
        v8f D = __builtin_amdgcn_wmma_f32_16x16x4_f32(
            false, A, false, Bm, (short)0, Cm, false, false);
#pragma unroll
        for (int r = 0; r < 8; ++r) {         // D layout: vgpr r -> row r + 8*half
            int row = half * 8 + r;
            dist[row * NPT + n] = D[r];
        }
    }
    __syncthreads();

    // Selection: each wave extracts 24 minima for 2 rows; vectorized b128 scans.
    for (int rr = 0; rr < 2; ++rr) {
        int row = wave * 2 + rr;
        const float4* rowp = (const float4*)(dist + row * NPT);
        for (int t = 0; t < NK; ++t) {
            float bv = 3.4e38f; int bi = 0x7FFFFFFF;
            for (int i = lane; i < NPT / 4; i += 32) {
                float4 v = rowp[i];
                int base = i * 4;              // per-lane indices increase -> '<'
                if (v.x < bv) { bv = v.x; bi = base; }
                if (v.y < bv) { bv = v.y; bi = base + 1; }
                if (v.z < bv) { bv = v.z; bi = base + 2; }
                if (v.w < bv) { bv = v.w; bi = base + 3; }
            }
#pragma unroll
            for (int off = 16; off > 0; off >>= 1) {
                float ov = __shfl_xor(bv, off, 32);
                int   oi = __shfl_xor(bi, off, 32);
                if (ov < bv || (ov == bv && oi < bi)) { bv = ov; bi = oi; }
            }
            if (lane == 0) {
                knn_idx[(b * NS + stile * 16 + row) * NK + t] = bi;
                dist[row * NPT + bi] = 3.4e38f;     // knock out, wave-local RAW
            }
            asm volatile("s_wait_dscnt 0" ::: "memory");
        }
    }
}

// ---------------- Kernel 4: zero per-batch accumulators --------------------
__global__ void zero_kernel(float* __restrict__ bsum) {
    if (threadIdx.x < NB) bsum[threadIdx.x] = 0.f;
}

// ------- Kernel 5: per-(b,s) sum of squared diffs; mean_k(diff)==0 ---------
__global__ __launch_bounds__(256) void ssq_kernel(const float* __restrict__ p,
                                                  const int* __restrict__ knn_idx,
                                                  float* __restrict__ bsum) {
    int gw   = blockIdx.x * 8 + (threadIdx.x >> 5);   // one wave per (b,s)
    int lane = threadIdx.x & 31;
    int b = gw >> 10;
    int s = gw & 1023;
    int idxs[NK];
#pragma unroll
    for (int k = 0; k < NK; ++k) idxs[k] = knn_idx[(b * NS + s) * NK + k];
    float ssq = 0.f;
    for (int cc = 0; cc < NC / 32; ++cc) {
        int c = lane + cc * 32;
        float v[NK]; float mean = 0.f;
#pragma unroll
        for (int k = 0; k < NK; ++k) {
            v[k] = p[((size_t)b * NPT + idxs[k]) * NC + c];
            mean += v[k];
        }
        mean *= (1.0f / NK);
#pragma unroll
        for (int k = 0; k < NK; ++k) { float d = v[k] - mean; ssq += d * d; }
    }
#pragma unroll
    for (int off = 16; off > 0; off >>= 1) ssq += __shfl_xor(ssq, off, 32);
    if (lane == 0) atomicAdd(&bsum[b], ssq);
}

// ---------------- Kernel 6: normalize + concat output ----------------------
__global__ __launch_bounds__(128) void final_kernel(const float* __restrict__ p,
                                                    const int* __restrict__ fps_idx,
                                                    const int* __restrict__ knn_idx,
                                                    const float* __restrict__ bsum,
                                                    const float* __restrict__ alpha,
                                                    const float* __restrict__ beta,
                                                    float* __restrict__ out) {
    int bs = blockIdx.x;                  // (b,s) pair
    int b  = bs >> 10;
    int c  = threadIdx.x;
    float stdv = sqrtf(bsum[b] / (float)((size_t)NS * NK * NC - 1));
    float inv  = 1.0f / (stdv + EPSV);
    float al = alpha[c], be = beta[c];
    int fidx = fps_idx[bs];
    float anchor = p[((size_t)b * NPT + fidx) * NC + c];
    float v[NK]; float mean = 0.f;
#pragma unroll
    for (int k = 0; k < NK; ++k) {
        v[k] = p[((size_t)b * NPT + knn_idx[bs * NK + k]) * NC + c];
        mean += v[k];
    }
    mean *= (1.0f / NK);
    float* o = out + (size_t)NB * NS * 3 + (size_t)bs * NK * 2 * NC;
#pragma unroll
    for (int k = 0; k < NK; ++k) {
        o[(size_t)k * 2 * NC + c]      = al * ((v[k] - mean) * inv) + be;
        o[(size_t)k * 2 * NC + NC + c] = anchor;
    }
}

extern "C" void kernel_launch(void* const* d_in, const int* in_sizes, int n_in,
                              void* d_out, int out_size, void* d_ws, size_t ws_size,
                              hipStream_t stream) {
    const float* xyz   = (const float*)d_in[0];
    const float* p     = (const float*)d_in[1];
    const float* alpha = (const float*)d_in[2];
    const float* beta  = (const float*)d_in[3];
    float* out = (float*)d_out;

    char* ws = (char*)d_ws;
    int*   fps_idx = (int*)ws;                                  // 16384 ints
    int*   knn_idx = (int*)(ws + 65536);                        // 393216 ints
    float* bsum    = (float*)(ws + 65536 + (size_t)NB * NS * NK * 4);

    (void)in_sizes; (void)n_in; (void)out_size; (void)ws_size;

    // allow 256 KB dynamic LDS for the distance panel
    static_cast<void>(hipFuncSetAttribute((const void*)knn_kernel,
                         hipFuncAttributeMaxDynamicSharedMemorySize,
                         16 * NPT * (int)sizeof(float)));

    fps_kernel<<<NB, 256, 0, stream>>>(xyz, fps_idx);
    gather_xyz_kernel<<<(NB * NS + 255) / 256, 256, 0, stream>>>(xyz, fps_idx, out);
    knn_kernel<<<NB * (NS / 16), 256, 16 * NPT * sizeof(float), stream>>>(xyz, fps_idx, knn_idx);
    zero_kernel<<<1, 32, 0, stream>>>(bsum);
    ssq_kernel<<<(NB * NS) / 8, 256, 0, stream>>>(p, knn_idx, bsum);
    final_kernel<<<NB * NS, 128, 0, stream>>>(p, fps_idx, knn_idx, bsum, alpha, beta, out);
}